// GINLinear_55594056679593
// MI455X (gfx1250) — compile-verified
//
#include <hip/hip_runtime.h>

// ---------------------------------------------------------------------------
// GIN layer for MI455X (gfx1250, wave32):
//   acc = h                      (init kernel)
//   acc[dst] += h[src] * mask    (wave-per-edge gather + hw fp32 atomics, L2-resident)
//   out = acc @ W^T              (fp32 WMMA 16x16x4, exact fp32 math)
// ---------------------------------------------------------------------------

#define D_FEAT 128

typedef float v8f __attribute__((ext_vector_type(8)));
typedef float v2f __attribute__((ext_vector_type(2)));

// ---- Stage 1: acc = h  (vectorized float4 copy; EPS==0 so (1+eps)h == h) ----
__global__ void gin_init_acc(const float4* __restrict__ h4,
                             float4* __restrict__ acc4, int n4) {
  int i = blockIdx.x * blockDim.x + threadIdx.x;
  if (i < n4) acc4[i] = h4[i];
}

// ---- Stage 2: one wave32 per edge; lane L owns features [4L, 4L+4) ----------
// Coalesced 512B gather of h[src], 4 native fp32 atomic adds into acc[dst].
__global__ void gin_scatter(const float* __restrict__ h,
                            const float* __restrict__ mask,
                            const int* __restrict__ src,
                            const int* __restrict__ dst,
                            float* __restrict__ acc, int n_edges) {
  const int wavesPerBlock = blockDim.x >> 5;
  const int e = blockIdx.x * wavesPerBlock + (threadIdx.x >> 5);
  if (e >= n_edges) return;
  const int lane = threadIdx.x & 31;

  const int   s = src[e];          // uniform across wave -> scalarized
  const int   d = dst[e];
  const float m = mask[e];

  const float4 v = *(const float4*)(h + (size_t)s * D_FEAT + lane * 4);
  float* p = acc + (size_t)d * D_FEAT + lane * 4;

  // Relaxed agent-scope fp32 atomics -> global_atomic_add_f32 (no return).
  __hip_atomic_fetch_add(p + 0, v.x * m, __ATOMIC_RELAXED, __HIP_MEMORY_SCOPE_AGENT);
  __hip_atomic_fetch_add(p + 1, v.y * m, __ATOMIC_RELAXED, __HIP_MEMORY_SCOPE_AGENT);
  __hip_atomic_fetch_add(p + 2, v.z * m, __ATOMIC_RELAXED, __HIP_MEMORY_SCOPE_AGENT);
  __hip_atomic_fetch_add(p + 3, v.w * m, __ATOMIC_RELAXED, __HIP_MEMORY_SCOPE_AGENT);
}

// ---- Stage 3: out[n][m] = sum_k acc[n][k] * W[m][k] via V_WMMA_F32_16X16X4_F32
// Block = 128 threads = 4 waves. Wave w computes rows [16*bx,16*bx+16) x
// out-features [64*by + 16*w, +16). K=128 walked in 32 steps of 4.
//
// fp32 A 16x4 layout (ISA 7.12.2): lanes 0-15 hold K={0,1} in {v0,v1},
// lanes 16-31 hold K={2,3}.  B 4x16 mirrors it (rows striped across lanes,
// second lane-half = K+2).  Since B[k][n] = W[n][k] (row-major W), both A and
// B reduce to the same float2 load: mat[(base + lane%16)*128 + k + (lane/16)*2].
__global__ void gin_gemm_wmma(const float* __restrict__ acc,
                              const float* __restrict__ W,
                              float* __restrict__ out, int n_nodes) {
  const int wave    = threadIdx.x >> 5;
  const int lane    = threadIdx.x & 31;
  const int l16     = lane & 15;
  const int halfsel = lane >> 4;            // 0: K+{0,1}, 1: K+{2,3}

  const int rowBase = blockIdx.x * 16;      // 16 output rows per block
  const int m0      = blockIdx.y * 64 + wave * 16;  // 16 out-features per wave
  if (rowBase >= n_nodes) return;

  const float* aPtr = acc + (size_t)(rowBase + l16) * D_FEAT + halfsel * 2;
  const float* bPtr = W   + (size_t)(m0      + l16) * D_FEAT + halfsel * 2;

  v8f c = {};
#pragma unroll
  for (int k = 0; k < D_FEAT; k += 4) {
    v2f a = *(const v2f*)(aPtr + k);
    v2f b = *(const v2f*)(bPtr + k);
    // 8 args: (neg_a, A, neg_b, B, c_mod, C, reuse_a, reuse_b)
    c = __builtin_amdgcn_wmma_f32_16x16x4_f32(false, a, false, b,
                                              (short)0, c, false, false);
  }

  // C/D layout: VGPR v -> M = v + 8*halfsel, N = l16. Coalesced 64B stores.
#pragma unroll
  for (int v = 0; v < 8; ++v) {
    const int row = rowBase + v + halfsel * 8;
    out[(size_t)row * D_FEAT + m0 + l16] = c[v];
  }
}

extern "C" void kernel_launch(void* const* d_in, const int* in_sizes, int n_in,
                              void* d_out, int out_size, void* d_ws, size_t ws_size,
                              hipStream_t stream) {
  const float* h    = (const float*)d_in[0];
  const float* mask = (const float*)d_in[1];
  const int*   src  = (const int*)d_in[2];
  const int*   dst  = (const int*)d_in[3];
  const float* W    = (const float*)d_in[4];
  float*       out  = (float*)d_out;

  const int n_nodes = in_sizes[0] / D_FEAT;   // 10000
  const int n_edges = in_sizes[2];            // 640000

  float* acc = (float*)d_ws;                  // n_nodes * 128 fp32 = 5.12 MB

  // 1) acc = h
  {
    int n4 = n_nodes * D_FEAT / 4;
    int blk = 256, grd = (n4 + blk - 1) / blk;
    gin_init_acc<<<grd, blk, 0, stream>>>((const float4*)h, (float4*)acc, n4);
  }
  // 2) scatter-add messages (8 waves / 8 edges per 256-thread block)
  {
    int blk = 256;
    int wavesPerBlock = blk / 32;
    int grd = (n_edges + wavesPerBlock - 1) / wavesPerBlock;
    gin_scatter<<<grd, blk, 0, stream>>>(h, mask, src, dst, acc, n_edges);
  }
  // 3) fp32 WMMA projection: grid (row tiles, feature half)
  {
    dim3 blk(128, 1, 1);
    dim3 grd((n_nodes + 15) / 16, D_FEAT / 64, 1);  // 625 x 2
    gin_gemm_wmma<<<grd, blk, 0, stream>>>(acc, W, out, n_nodes);
  }
}